// JAXModel_51445118271627
// MI455X (gfx1250) — compile-verified
//
#include <hip/hip_runtime.h>

// Problem constants (match reference setup_inputs)
#define BN 512
#define TN 64
#define SN 128
#define ON 32
#define DT 0.05f
#define HALF_LOG2PI 0.91893853320467274f

// padded LDS strides (stride 129 floats -> bank = (row+col) mod 64, conflict free)
#define ST  129   // K matrix stride
#define STX 129   // X (state history) stride
#define STC 129   // C matrix stride

typedef float v2f __attribute__((ext_vector_type(2)));
typedef float v8f __attribute__((ext_vector_type(8)));

// LDS float counts
// Km : SN*ST      (128x129)  = 16512
// X  : TN*STX     (64x129)   =  8256
// Cls: ON*STC     (32x129)   =  4128
// xa : SN, xb : SN, red : 128, sP : 16
#define LDS_FLOATS (SN*ST + TN*STX + ON*STC + SN + SN + 128 + 16)

extern "C" __global__ __launch_bounds__(128)
void batch_llh_kernel(const float* __restrict__ p,
                      const float* __restrict__ A,
                      const float* __restrict__ Cm,
                      const float* __restrict__ dvec,
                      const float* __restrict__ sigp,
                      const float* __restrict__ my,
                      const int*   __restrict__ pscale,
                      float* __restrict__ partial)
{
    extern __shared__ float lds[];
    float* Km  = lds;                    // [SN][ST]
    float* X   = Km  + SN * ST;          // [TN][STX]
    float* Cls = X   + TN * STX;         // [ON][STC]
    float* xa  = Cls + ON * STC;         // [SN]
    float* xb  = xa  + SN;               // [SN]
    float* red = xb  + SN;               // [128]
    float* sP  = red + 128;              // pivot broadcast

    const int tid = threadIdx.x;         // 0..127, 4 waves of 32
    const int b   = blockIdx.x;          // batch index

    // ---- load C (coalesced) into LDS -------------------------------------
    for (int idx = tid; idx < ON * SN; idx += 128) {
        int o = idx >> 7, s = idx & 127;
        Cls[o * STC + s] = Cm[idx];
    }

    // ---- per-state parameter transforms (thread tid owns state tid) ------
    float rdecay;
    {
        float pv = p[b * SN + tid];
        int   sc = pscale[tid];
        float ps = (sc == 0) ? pv : ((sc == 1) ? expf(pv) : exp10f(pv));
        // softplus (stable) and sigmoid
        rdecay   = fmaxf(ps, 0.0f) + log1pf(expf(-fabsf(ps)));
        xa[tid]  = 1.0f / (1.0f + expf(-ps));      // x0
    }

    // ---- build K = I - DT*A/sqrt(S) + DT*diag(r) (coalesced over columns)
    const float coef = -DT / sqrtf((float)SN);
    for (int row = 0; row < SN; ++row) {
        Km[row * ST + tid] = coef * A[row * SN + tid];
    }
    // diagonal element (row tid, col tid) was written by this same thread
    Km[tid * ST + tid] += 1.0f + DT * rdecay;
    __syncthreads();

    // ---- in-place Gauss-Jordan inversion (no pivoting: diag dominant) ----
    for (int k = 0; k < SN; ++k) {
        if (tid == 0) sP[0] = 1.0f / Km[k * ST + k];
        __syncthreads();
        const float piv = sP[0];
        if (tid != k) Km[k * ST + tid] *= piv;     // scale row k (col-parallel)
        else          Km[k * ST + k]    = piv;
        __syncthreads();
        if (tid != k) {                            // thread owns row tid
            float*       rowi = &Km[tid * ST];
            const float* rowk = &Km[k   * ST];
            const float  f    = rowi[k];
            #pragma unroll 8
            for (int j = 0; j < SN; ++j) rowi[j] -= f * rowk[j];
            rowi[k] = -f * piv;                    // fix j==k slot
        }
        __syncthreads();
    }
    // Km now holds K^{-1}

    // ---- implicit-Euler recurrence: x_{t+1} = K^{-1} x_t, store into X ---
    float* cur = xa;
    float* nxt = xb;
    for (int t = 0; t < TN; ++t) {
        const float* rowi = &Km[tid * ST];
        float s = 0.0f;
        #pragma unroll 8
        for (int j = 0; j < SN; ++j) s = fmaf(rowi[j], cur[j], s);
        nxt[tid]           = s;
        X[t * STX + tid]   = s;
        __syncthreads();
        float* tmp = cur; cur = nxt; nxt = tmp;
    }

    // ---- Y[64x32] = X[64x128] @ C^T via V_WMMA_F32_16X16X4_F32 -----------
    // tiles: mt in 0..3 (time), nt in 0..1 (obs); wave w does tiles w, w+4
    const int wave = tid >> 5;
    const int lane = tid & 31;
    const int l    = lane & 15;
    const int half = lane >> 4;

    float lsum = 0.0f;
    for (int tl = wave; tl < 8; tl += 4) {
        const int mt   = tl & 3;
        const int nt   = tl >> 2;
        const int arow = mt * 16 + l;              // time row within X
        const int ocol = nt * 16 + l;              // observable column
        v8f acc = {};
        for (int kk = 0; kk < 32; ++kk) {
            const int kb = kk * 4 + half * 2;      // K split: lanes0-15 K=0,1; 16-31 K=2,3
            v2f a, bb;
            a.x  = X[arow * STX + kb];
            a.y  = X[arow * STX + kb + 1];
            bb.x = Cls[ocol * STC + kb];           // B[k][n] = C[n][k]
            bb.y = Cls[ocol * STC + kb + 1];
            acc = __builtin_amdgcn_wmma_f32_16x16x4_f32(
                false, a, false, bb, (short)0, acc, false, false);
        }
        // Gaussian NLL accumulation for this tile
        const float sig   = sigp[ocol];            // log(sigy) == sigma_p
        const float isigy = expf(-sig);
        const float dv    = dvec[ocol];
        #pragma unroll
        for (int vi = 0; vi < 8; ++vi) {
            const int t = mt * 16 + vi + half * 8; // D layout: VGPR vi -> M = vi (+8 hi half)
            const float y = acc[vi] + dv;
            const float m = my[((size_t)b * TN + t) * ON + ocol];
            const float e = (y - m) * isigy;
            lsum += 0.5f * e * e + sig + HALF_LOG2PI;
        }
    }

    // ---- block reduction -> per-batch partial Jy sum ---------------------
    red[tid] = lsum;
    __syncthreads();
    for (int off = 64; off > 0; off >>= 1) {
        if (tid < off) red[tid] += red[tid + off];
        __syncthreads();
    }
    if (tid == 0) partial[b] = red[0];
}

extern "C" __global__ __launch_bounds__(256)
void reduce_llh_kernel(const float* __restrict__ partial, float* __restrict__ out)
{
    __shared__ float red[256];
    const int tid = threadIdx.x;
    red[tid] = partial[tid] + partial[tid + 256];
    __syncthreads();
    for (int off = 128; off > 0; off >>= 1) {
        if (tid < off) red[tid] += red[tid + off];
        __syncthreads();
    }
    if (tid == 0) out[0] = -red[0];               // llh = -sum(Jy)
}

extern "C" void kernel_launch(void* const* d_in, const int* in_sizes, int n_in,
                              void* d_out, int out_size, void* d_ws, size_t ws_size,
                              hipStream_t stream)
{
    const float* p      = (const float*)d_in[0];   // [512,128]
    const float* A      = (const float*)d_in[1];   // [128,128]
    const float* C      = (const float*)d_in[2];   // [32,128]
    const float* dvec   = (const float*)d_in[3];   // [32]
    const float* sigp   = (const float*)d_in[4];   // [32]
    const float* my     = (const float*)d_in[5];   // [512,64,32]
    const int*   pscale = (const int*)d_in[6];     // [128]
    float* partial = (float*)d_ws;                 // 512 floats
    float* out     = (float*)d_out;                // scalar

    const size_t ldsBytes = (size_t)LDS_FLOATS * sizeof(float); // ~114.4 KB < 320 KB WGP LDS

    batch_llh_kernel<<<BN, 128, ldsBytes, stream>>>(p, A, C, dvec, sigp, my, pscale, partial);
    reduce_llh_kernel<<<1, 256, 0, stream>>>(partial, out);
}